// AdaptedGNN_77670188580918
// MI455X (gfx1250) — compile-verified
//
#include <hip/hip_runtime.h>
#include <hip/hip_bf16.h>

// AdaptedGNN affinity kernel for MI455X (gfx1250).
//
// Reference: affinity[r][i] = (f[i,2]+f[i,4]+eps) / (||robot_r - f[i,0:2]|| + eps)
// for r in {0,1}, i in [0, M). Inputs d_in[1]/d_in[2] (edge data) are unused
// by the reference and therefore ignored here.
//
// Roofline: ~160 MB input + 64 MB output = 224 MB -> ~9.6 us @ 23.3 TB/s HBM.
// ~0.16 GFLOP of math -> purely bandwidth bound. No matrix structure -> WMMA
// not applicable; optimize the data-movement path instead (wave32 blocks,
// aligned wide vmem, non-temporal temporal-hints, global_prefetch_b8).

static constexpr float KEPS = 1e-6f;

typedef float f2v __attribute__((ext_vector_type(2)));
typedef float f4v __attribute__((ext_vector_type(4)));

// Vectorized kernel: one thread = 4 frontier rows (20 floats).
// Frontier base = nf + 10 floats -> byte offset 40 + 80*g: 8-byte aligned,
// so the 20 floats are exactly 10 aligned b64 (non-temporal) loads.
// Requires M % 4 == 0 so both output-row b128 stores stay 16B aligned.
__global__ __launch_bounds__(256) void affinity_vec4_kernel(
    const float* __restrict__ nf, float* __restrict__ out,
    long long nGroups, long long M)
{
    // Robot positions: uniform address loads, hit in cache for every wave.
    const float rx0 = nf[0], ry0 = nf[1];
    const float rx1 = nf[5], ry1 = nf[6];
    const float* __restrict__ fr = nf + 10;

    long long g      = (long long)blockIdx.x * blockDim.x + threadIdx.x;
    long long stride = (long long)gridDim.x * blockDim.x;

    for (; g < nGroups; g += stride) {
        const f2v* p = (const f2v*)(fr + 20ll * g);

        // gfx1250 prefetch path (global_prefetch_b8); speculative, OOB-safe.
        __builtin_prefetch(fr + 20ll * g + 2048, 0, 1);

        // 10 aligned non-temporal b64 loads = 4 frontier rows.
        f2v v0 = __builtin_nontemporal_load(p + 0);
        f2v v1 = __builtin_nontemporal_load(p + 1);
        f2v v2 = __builtin_nontemporal_load(p + 2);
        f2v v3 = __builtin_nontemporal_load(p + 3);
        f2v v4 = __builtin_nontemporal_load(p + 4);
        f2v v5 = __builtin_nontemporal_load(p + 5);
        f2v v6 = __builtin_nontemporal_load(p + 6);
        f2v v7 = __builtin_nontemporal_load(p + 7);
        f2v v8 = __builtin_nontemporal_load(p + 8);
        f2v v9 = __builtin_nontemporal_load(p + 9);

        // Group-local float k = row j*5 + c:
        //   row0: 0..4  -> v0.x v0.y v1.x (v1.y) v2.x
        //   row1: 5..9  -> v2.y v3.x v3.y (v4.x) v4.y
        //   row2: 10..14-> v5.x v5.y v6.x (v6.y) v7.x
        //   row3: 15..19-> v7.y v8.x v8.y (v9.x) v9.y
        float fx[4] = { v0.x, v2.y, v5.x, v7.y };
        float fy[4] = { v0.y, v3.x, v5.y, v8.x };
        float gn[4] = { v1.x + v2.x + KEPS,
                        v3.y + v4.y + KEPS,
                        v6.x + v7.x + KEPS,
                        v8.y + v9.y + KEPS };

        f4v o0, o1;
#pragma unroll
        for (int j = 0; j < 4; ++j) {
            float dx0 = rx0 - fx[j], dy0 = ry0 - fy[j];
            float dx1 = rx1 - fx[j], dy1 = ry1 - fy[j];
            float d0 = __builtin_amdgcn_sqrtf(__builtin_fmaf(dx0, dx0, dy0 * dy0)) + KEPS;
            float d1 = __builtin_amdgcn_sqrtf(__builtin_fmaf(dx1, dx1, dy1 * dy1)) + KEPS;
            o0[j] = gn[j] * __builtin_amdgcn_rcpf(d0);
            o1[j] = gn[j] * __builtin_amdgcn_rcpf(d1);
        }

        // Two aligned non-temporal b128 stores (one per robot row).
        __builtin_nontemporal_store(o0, (f4v*)(out + 4ll * g));
        __builtin_nontemporal_store(o1, (f4v*)(out + M + 4ll * g));
    }
}

// Scalar fallback (used only if M % 4 != 0; still coalesced).
__global__ __launch_bounds__(256) void affinity_scalar_kernel(
    const float* __restrict__ nf, float* __restrict__ out, long long M)
{
    const float rx0 = nf[0], ry0 = nf[1];
    const float rx1 = nf[5], ry1 = nf[6];

    long long i      = (long long)blockIdx.x * blockDim.x + threadIdx.x;
    long long stride = (long long)gridDim.x * blockDim.x;

    for (; i < M; i += stride) {
        const float* row = nf + 10ll + 5ll * i;
        float x  = row[0];
        float y  = row[1];
        float gn = row[2] + row[4] + KEPS;

        float dx0 = rx0 - x, dy0 = ry0 - y;
        float dx1 = rx1 - x, dy1 = ry1 - y;
        float d0 = __builtin_amdgcn_sqrtf(__builtin_fmaf(dx0, dx0, dy0 * dy0)) + KEPS;
        float d1 = __builtin_amdgcn_sqrtf(__builtin_fmaf(dx1, dx1, dy1 * dy1)) + KEPS;

        __builtin_nontemporal_store(gn * __builtin_amdgcn_rcpf(d0), out + i);
        __builtin_nontemporal_store(gn * __builtin_amdgcn_rcpf(d1), out + M + i);
    }
}

extern "C" void kernel_launch(void* const* d_in, const int* in_sizes, int n_in,
                              void* d_out, int out_size, void* d_ws, size_t ws_size,
                              hipStream_t stream) {
    (void)n_in; (void)out_size; (void)d_ws; (void)ws_size;

    const float* nf  = (const float*)d_in[0];   // node_features (N, 5) fp32
    float*       out = (float*)d_out;           // affinity (2, M) fp32

    const long long N = (long long)in_sizes[0] / 5ll;
    const long long M = N - 2ll;                 // 8,000,000 for this shape
    if (M <= 0) return;

    const int threads = 256;                     // 8 wave32 per block

    if ((M & 3ll) == 0ll) {
        const long long nGroups = M >> 2;        // 2,000,000 groups of 4 rows
        long long blocks = (nGroups + threads - 1) / threads;
        if (blocks > 1048576ll) blocks = 1048576ll;  // grid-stride covers rest
        affinity_vec4_kernel<<<(int)blocks, threads, 0, stream>>>(nf, out, nGroups, M);
    } else {
        long long blocks = (M + threads - 1) / threads;
        if (blocks > 1048576ll) blocks = 1048576ll;
        affinity_scalar_kernel<<<(int)blocks, threads, 0, stream>>>(nf, out, M);
    }
}